// filter_deep_amp_6751688589399
// MI455X (gfx1250) — compile-verified
//
#include <hip/hip_runtime.h>
#include <cmath>

// ---------------------------------------------------------------------------
// filter_deep_amp pipeline for gfx1250 (MI455X).
// Conv layers: implicit GEMM on V_WMMA_F32_16X16X4_F32 (fp32 matrix path).
//   M = 16 output pixels (x-contiguous), N = co padded to 16, K = ci*k*k.
// Activations live in zero-halo padded planes (288x288, image at (16,16)),
// so conv zero-padding comes free from memory: a single branch-free,
// check-free load path for every block. Each wave keeps 8 independent
// accumulator tiles (8 output rows) -> 8 independent WMMA chains;
// B-fragments come from an LDS weight panel once per K-chunk, reused 8x.
// Global accesses use scalar base + 32-bit lane offset + immediate (GVS).
// ---------------------------------------------------------------------------

typedef __attribute__((ext_vector_type(2))) float v2f;
typedef __attribute__((ext_vector_type(8))) float v8f;

#define IMG   256
#define IMG2  (IMG * IMG)
#define BATCH 32
#define HALO  16
#define PS    (IMG + 2 * HALO)        // 288: padded row stride
#define PSZ   (PS * PS)               // 82944: padded plane size (floats)
#define PSZ4  (PSZ / 4)               // 20736: plane size in float4
#define NSLOT 16                      // channel slots per batch in a buffer
#define EPSV      1e-5f
#define ELU_ADDV  1.00001f            // 1.0 + 1e-5

// Padded-plane float index of (slot s of batch b, image row y, image col x):
//   (b*NSLOT + s)*PSZ + (HALO+y)*PS + (HALO+x)

template <int CI, int CO, int KSZ, int PAD, int DIL>
__global__ __launch_bounds__(256) void conv_bn_elu_wmma(
    const float* __restrict__ in,   // padded: [B][NSLOT][288][288]
    const float* __restrict__ w,    // [CO][CI][KSZ][KSZ]
    const float* __restrict__ bng, const float* __restrict__ bnb,
    const float* __restrict__ bnm, const float* __restrict__ bnv,
    float* __restrict__ out)        // padded: [B][NSLOT][288][288]
{
    constexpr int K    = CI * KSZ * KSZ;
    constexpr int KP4  = (KSZ + 3) & ~3;   // filter row padded to K-chunks of 4
    constexpr int NQ   = KP4 / 4;          // K-chunks per filter row
    constexpr int ROWS = 8;                // output rows per wave
    extern __shared__ float wlds[];        // [CI*KSZ][KP4][16]

    // Cooperative zero-padded weight panel load (compile-time divisors).
    constexpr int TOT = CI * KSZ * KP4 * 16;
    for (int idx = threadIdx.x; idx < TOT; idx += 256) {
        int nn  = idx & 15;
        int kk  = idx >> 4;
        int row = kk / KP4;
        int kx  = kk - row * KP4;
        float val = 0.0f;
        if (kx < KSZ && nn < CO) val = w[nn * K + row * KSZ + kx];
        wlds[idx] = val;
    }
    __syncthreads();

    const int lane = threadIdx.x & 31;
    const int wave = threadIdx.x >> 5;
    const int ox0  = (blockIdx.x * 8 + wave) * 16;
    const int oyb  = blockIdx.y * ROWS;
    const int bb   = blockIdx.z;

    const int h = lane >> 4;        // lane half: selects K pair of the chunk
    const int m = lane & 15;        // A-matrix row = pixel within tile
    const int n = m;                // B/D column  = output channel

    const float* inb  = in + (size_t)bb * NSLOT * PSZ;   // uniform
    const int    ox_m = ox0 + m - PAD;                   // lane-varying, kx=0

    v8f zero = {};
    v8f acc[ROWS];
    #pragma unroll
    for (int r = 0; r < ROWS; ++r) acc[r] = zero;

    // Single check-free path: halo zeros supply the conv padding.
    for (int c = 0; c < CI; ++c) {
        const float* cb = inb + (size_t)c * PSZ;                       // uniform
        for (int ky = 0; ky < KSZ; ++ky) {
            const float* rowb = cb + (HALO + oyb - PAD + ky * DIL) * PS + HALO;
            const int kbase = (c * KSZ + ky) * KP4;
            #pragma unroll
            for (int q = 0; q < NQ; ++q) {
                const int kx0 = q * 4 + 2 * h;    // 32-bit A 16x4 layout
                v2f b;
                b.x = wlds[(kbase + kx0) * 16 + n];
                b.y = wlds[(kbase + kx0 + 1) * 16 + n];
                const int xo = ox_m + kx0 * DIL;  // lane offset, ry-invariant
                #pragma unroll
                for (int ry = 0; ry < ROWS; ++ry) {
                    v2f a;
                    a.x = rowb[ry * PS + xo];        // saddr + imm ry*1152
                    a.y = rowb[ry * PS + xo + DIL];  // saddr + imm +4*DIL
                    acc[ry] = __builtin_amdgcn_wmma_f32_16x16x4_f32(
                        false, a, false, b, (short)0, acc[ry], false, false);
                }
            }
        }
    }

    // Epilogue. D layout: VGPR r, lane l -> M = (l>>4)*8 + r, N = l&15.
    if (n < CO) {
        const float scl = bng[n] * rsqrtf(bnv[n] + EPSV);
        const float mu  = bnm[n];
        const float bet = bnb[n];
        float* ob = out + (size_t)bb * NSLOT * PSZ
                        + (size_t)(HALO + oyb) * PS + HALO;   // uniform
        const int oo = n * PSZ + ox0 + h * 8;                 // lane offset
        #pragma unroll
        for (int ry = 0; ry < ROWS; ++ry) {
            #pragma unroll
            for (int r = 0; r < 8; ++r) {
                float y = acc[ry][r];
                y = (y - mu) * scl + bet;          // batch norm
                y = (y > 0.0f) ? y : expm1f(y);    // ELU (alpha=1)
                ob[ry * PS + oo + r] = y + ELU_ADDV;
            }
        }
    }
}

// ---------------------------------------------------------------------------
// Zero-fill workspace (halo zeros = conv padding). float4 stores.
// ---------------------------------------------------------------------------
__global__ void zero_ws(float4* __restrict__ p, int n4)
{
    int idx = blockIdx.x * blockDim.x + threadIdx.x;
    if (idx < n4) {
        float4 z; z.x = 0.f; z.y = 0.f; z.z = 0.f; z.w = 0.f;
        p[idx] = z;
    }
}

// ---------------------------------------------------------------------------
// Locally-connected block (float4, bandwidth bound), padded src/dst planes:
//   out[b,0,y,x] = relu(bn( sum_c in[b,c,y,x] * w[0,c,y,x] ))
// Thread = one float4 of one image row. 64 float4 per row.
// ---------------------------------------------------------------------------
__global__ void lc_bn_relu(const float* __restrict__ in, const float* __restrict__ w,
                           const float* __restrict__ g, const float* __restrict__ b,
                           const float* __restrict__ m, const float* __restrict__ v,
                           float* __restrict__ out, int ci)
{
    int idx = blockIdx.x * blockDim.x + threadIdx.x;
    if (idx >= BATCH * IMG * 64) return;
    int bb  = idx / (IMG * 64);
    int rem = idx - bb * (IMG * 64);
    int y   = rem >> 6;
    int x4  = rem & 63;
    // float4 index of (slot, y, x4) in padded plane: HALO offsets are /4 exact.
    const int rowoff4 = (HALO + y) * (PS / 4) + (HALO / 4) + x4;
    const float4* in4 = reinterpret_cast<const float4*>(in);
    const float4* w4  = reinterpret_cast<const float4*>(w);
    float sx = 0.f, sy = 0.f, sz = 0.f, sw = 0.f;
    for (int c = 0; c < ci; ++c) {
        float4 av = in4[(size_t)(bb * NSLOT + c) * PSZ4 + rowoff4];
        float4 wv = w4[(size_t)c * (IMG2 / 4) + (y << 6) + x4];  // unpadded w
        sx += av.x * wv.x; sy += av.y * wv.y;
        sz += av.z * wv.z; sw += av.w * wv.w;
    }
    const float scl = g[0] * rsqrtf(v[0] + EPSV);
    const float mu = m[0], bet = b[0];
    float4 o;
    o.x = (sx - mu) * scl + bet; o.x = o.x > 0.f ? o.x : 0.f;
    o.y = (sy - mu) * scl + bet; o.y = o.y > 0.f ? o.y : 0.f;
    o.z = (sz - mu) * scl + bet; o.z = o.z > 0.f ? o.z : 0.f;
    o.w = (sw - mu) * scl + bet; o.w = o.w > 0.f ? o.w : 0.f;
    reinterpret_cast<float4*>(out)[(size_t)(bb * NSLOT) * PSZ4 + rowoff4] = o;
}

// ---------------------------------------------------------------------------
// Extract channel 0 of x (B,2,256,256) into slot 0 of padded buffer.
// ---------------------------------------------------------------------------
__global__ void extract_ch0(const float* __restrict__ x, float* __restrict__ out)
{
    int idx = blockIdx.x * blockDim.x + threadIdx.x;
    if (idx >= BATCH * IMG * 64) return;
    int bb  = idx / (IMG * 64);
    int rem = idx - bb * (IMG * 64);
    int y   = rem >> 6;
    int x4  = rem & 63;
    float4 val = reinterpret_cast<const float4*>(x)
                    [(size_t)(bb * 2) * (IMG2 / 4) + (y << 6) + x4];
    reinterpret_cast<float4*>(out)
        [(size_t)(bb * NSLOT) * PSZ4 + (HALO + y) * (PS / 4) + (HALO / 4) + x4] = val;
}

// ---------------------------------------------------------------------------
// Point-symmetrize + final select (gather formulation, race-free):
// dest (di,dj) pulls from (si,sj)=((-di)%N,(-dj)%N) iff (si,sj) in
//   S = { si>sj and not(sj==0 and si>N/2) } U { si==sj and si>=N/2 }.
// Then out = where(x_ch0 != 0, x_ch0, sym). amp is a padded plane (slot 0).
// ---------------------------------------------------------------------------
__global__ void sym_select(const float* __restrict__ amp, const float* __restrict__ x,
                           float* __restrict__ out)
{
    int idx = blockIdx.x * blockDim.x + threadIdx.x;
    if (idx >= BATCH * IMG2) return;
    int bb  = idx >> 16;
    int pix = idx & (IMG2 - 1);
    int di  = pix >> 8;
    int dj  = pix & (IMG - 1);
    int si  = (IMG - di) & (IMG - 1);
    int sj  = (IMG - dj) & (IMG - 1);
    bool inset = (si > sj && !(sj == 0 && si > IMG / 2)) ||
                 (si == sj && si >= IMG / 2);
    const float* ab = amp + (size_t)(bb * NSLOT) * PSZ + (size_t)HALO * PS + HALO;
    float val = inset ? ab[si * PS + sj] : ab[di * PS + dj];
    float inp = x[((size_t)bb * 2) * IMG2 + pix];
    out[idx] = (inp != 0.0f) ? inp : val;
}

// ---------------------------------------------------------------------------
// Host-side templated launcher: p = &d_in[1+5*i] -> (w, g, b, m, v)
// ---------------------------------------------------------------------------
template <int CI, int CO, int KSZ, int PAD, int DIL>
static void launch_conv(void* const* p, const float* src, float* dst,
                        hipStream_t stream)
{
    constexpr int KP4 = (KSZ + 3) & ~3;
    size_t shmem = (size_t)CI * KSZ * KP4 * 16 * sizeof(float); // max ~174KB < 320KB
    dim3 grid(2, IMG / 8, BATCH);   // 2 blk * 8 waves = 16 x-tiles; 8 rows/block
    conv_bn_elu_wmma<CI, CO, KSZ, PAD, DIL><<<grid, 256, shmem, stream>>>(
        src, (const float*)p[0], (const float*)p[1], (const float*)p[2],
        (const float*)p[3], (const float*)p[4], dst);
}

extern "C" void kernel_launch(void* const* d_in, const int* in_sizes, int n_in,
                              void* d_out, int out_size, void* d_ws, size_t ws_size,
                              hipStream_t stream)
{
    (void)in_sizes; (void)n_in; (void)out_size; (void)ws_size;

    const float* x = (const float*)d_in[0];
    const size_t BUFSZ = (size_t)BATCH * NSLOT * PSZ;   // floats per buffer
    float* buf0 = (float*)d_ws;
    float* buf1 = buf0 + BUFSZ;

    const int ROW_GRID = (BATCH * IMG * 64 + 255) / 256;  // float4-per-row kernels
    const int ELT_GRID = (BATCH * IMG2 + 255) / 256;
    const int Z4       = (int)(2 * BUFSZ / 4);
    const int Z_GRID   = (Z4 + 255) / 256;

    // Zero halos (and everything else) so padded loads read zeros.
    zero_ws<<<Z_GRID, 256, 0, stream>>>((float4*)d_ws, Z4);

    auto lc = [&](int j, const float* src, float* dst, int ci) {
        int base = 51 + 5 * j;
        lc_bn_relu<<<ROW_GRID, 256, 0, stream>>>(
            src, (const float*)d_in[base],
            (const float*)d_in[base + 1], (const float*)d_in[base + 2],
            (const float*)d_in[base + 3], (const float*)d_in[base + 4], dst, ci);
    };

    extract_ch0<<<ROW_GRID, 256, 0, stream>>>(x, buf0);

    // Group 0: conv0..2 + lc0                 (ci, co, k, pad, dil)
    launch_conv<1, 4, 23, 11, 1>(&d_in[1],  buf0, buf1, stream);
    launch_conv<4, 8, 21, 10, 1>(&d_in[6],  buf1, buf0, stream);
    launch_conv<8, 12, 17, 8, 1>(&d_in[11], buf0, buf1, stream);
    lc(0, buf1, buf0, 12);

    // Group 1: conv3..6 + lc1
    launch_conv<1, 4, 5, 4, 2>(&d_in[16],   buf0, buf1, stream);
    launch_conv<4, 8, 5, 2, 1>(&d_in[21],   buf1, buf0, stream);
    launch_conv<8, 12, 3, 2, 2>(&d_in[26],  buf0, buf1, stream);
    launch_conv<12, 16, 3, 1, 1>(&d_in[31], buf1, buf0, stream);
    lc(1, buf0, buf1, 16);

    // Group 2: conv7..9 + lc2
    launch_conv<1, 4, 3, 1, 1>(&d_in[36],   buf1, buf0, stream);
    launch_conv<4, 8, 3, 1, 1>(&d_in[41],   buf0, buf1, stream);
    launch_conv<8, 12, 3, 2, 2>(&d_in[46],  buf1, buf0, stream);
    lc(2, buf0, buf1, 12);

    sym_select<<<ELT_GRID, 256, 0, stream>>>(buf1, x, (float*)d_out);
}